// GNN_5866925326819
// MI455X (gfx1250) — compile-verified
//
#include <hip/hip_runtime.h>

// CDNA5 / gfx1250: wave32, WMMA 16x16 tiles, split wait counters.
typedef __attribute__((ext_vector_type(2))) float v2f;
typedef __attribute__((ext_vector_type(8))) float v8f;

// ---------------------------------------------------------------------------
// Edge message kernel:  agg[dst] += ReLU([h[src] | edge_attr] @ W + b)
// One wave per 16-edge tile, 4 waves (128 threads) per block.
// GEMM runs on V_WMMA_F32_16X16X4_F32 (full fp32 precision).
// ---------------------------------------------------------------------------
template<int HIN, int NOUT>
__global__ __launch_bounds__(128)
void msg_kernel(const float* __restrict__ h, const float* __restrict__ ea,
                const long long* __restrict__ src, const long long* __restrict__ dst,
                const float* __restrict__ W, const float* __restrict__ bias,
                float* __restrict__ agg, int ntiles)
{
    constexpr int K  = HIN + 4;           // 12 or 68, both multiples of 4
    constexpr int KP = K + 4;             // LDS row stride (multiple of 4 floats)
    constexpr int KC = K / 4;             // number of K=4 WMMA steps
    constexpr int NP = (NOUT < 16) ? 16 : NOUT;
    constexpr int NC = NP / 16;           // N chunks of 16

    __shared__ v2f   sW[(K / 2) * NP];    // (k-pair, n) -> {W[2k][n], W[2k+1][n]}
    __shared__ float sA[4][16 * KP];      // per-wave 16 x K A-tile
    __shared__ int   sDst[4][16];

    // cooperative preload of W into LDS (pair layout -> B frag = one ds_load_b64)
    for (int idx = threadIdx.x; idx < (K / 2) * NP; idx += 128) {
        const int k2 = idx / NP;
        const int n  = idx % NP;
        v2f w;
        w.x = (n < NOUT) ? W[(2 * k2    ) * NOUT + n] : 0.f;
        w.y = (n < NOUT) ? W[(2 * k2 + 1) * NOUT + n] : 0.f;
        sW[idx] = w;
    }
    __syncthreads();

    const int wave = threadIdx.x >> 5;
    const int lane = threadIdx.x & 31;
    const int m    = lane & 15;           // row within 16-row tile
    const int kg   = lane >> 4;           // k sub-group (0/1) per WMMA layout

    const int tile = blockIdx.x * 4 + wave;
    if (tile >= ntiles) return;           // uniform per wave
    const int tbase = tile * 16;

    // ---- stage A = [h[src] | edge_attr] rows into LDS (coalesced float4) ----
    {
        const long long sidx = src[tbase + m];
        const float4* hr = (const float4*)(h + (size_t)sidx * HIN);
        float4* ar = (float4*)&sA[wave][m * KP];
        if constexpr (HIN == 64) {
            #pragma unroll
            for (int i = 0; i < 8; ++i) ar[kg * 8 + i] = hr[kg * 8 + i];
        } else {                          // HIN == 8
            if (kg == 0) { ar[0] = hr[0]; ar[1] = hr[1]; }
        }
        if (kg == 0) {
            ((float4*)&sA[wave][m * KP + HIN])[0] = ((const float4*)ea)[tbase + m];
            sDst[wave][m] = (int)dst[tbase + m];
        }
    }
    // per-wave LDS visibility: CDNA5 split DS counter
    asm volatile("s_wait_dscnt 0" ::: "memory");

    // ---- [16 x K] * [K x NP] via v_wmma_f32_16x16x4_f32 ----
    v8f acc[NC];
    const v8f zacc = {0.f, 0.f, 0.f, 0.f, 0.f, 0.f, 0.f, 0.f};
    #pragma unroll
    for (int nc = 0; nc < NC; ++nc) acc[nc] = zacc;

    #pragma unroll 4
    for (int kc = 0; kc < KC; ++kc) {
        const int kk = kc * 4 + kg * 2;
        v2f a = *(const v2f*)&sA[wave][m * KP + kk];
        #pragma unroll
        for (int nc = 0; nc < NC; ++nc) {
            v2f b = sW[(kk >> 1) * NP + nc * 16 + m];
            acc[nc] = __builtin_amdgcn_wmma_f32_16x16x4_f32(
                false, a, false, b, (short)0, acc[nc], false, false);
        }
    }

    // ---- bias + ReLU + scatter-add into agg[dst] (L2-resident atomics) ----
    #pragma unroll
    for (int nc = 0; nc < NC; ++nc) {
        const int n  = nc * 16 + m;
        const float bv = (n < NOUT) ? bias[n] : 0.f;
        #pragma unroll
        for (int v = 0; v < 8; ++v) {
            const int row = v + kg * 8;   // C/D layout: VGPR v -> M=v (+8 for hi half)
            float val = fmaxf(acc[nc][v] + bv, 0.f);
            if (n < NOUT) {
                const int d = sDst[wave][row];
                atomicAdd(&agg[(size_t)d * NOUT + n], val);
            }
        }
    }
}

// ---------------------------------------------------------------------------
// Node update kernel:  out = ReLU([h | agg] @ W + b)   (dense rows, no atomics)
// ---------------------------------------------------------------------------
template<int HPREV, int AGGD, int NOUT>
__global__ __launch_bounds__(128)
void upd_kernel(const float* __restrict__ h, const float* __restrict__ agg,
                const float* __restrict__ W, const float* __restrict__ bias,
                float* __restrict__ out, int ntiles)
{
    constexpr int K  = HPREV + AGGD;      // 72 / 128 / 68
    constexpr int KP = K + 4;
    constexpr int KC = K / 4;
    constexpr int NP = (NOUT < 16) ? 16 : NOUT;
    constexpr int NC = NP / 16;

    __shared__ v2f   sW[(K / 2) * NP];
    __shared__ float sA[4][16 * KP];

    for (int idx = threadIdx.x; idx < (K / 2) * NP; idx += 128) {
        const int k2 = idx / NP;
        const int n  = idx % NP;
        v2f w;
        w.x = (n < NOUT) ? W[(2 * k2    ) * NOUT + n] : 0.f;
        w.y = (n < NOUT) ? W[(2 * k2 + 1) * NOUT + n] : 0.f;
        sW[idx] = w;
    }
    __syncthreads();

    const int wave = threadIdx.x >> 5;
    const int lane = threadIdx.x & 31;
    const int m    = lane & 15;
    const int kg   = lane >> 4;

    const int tile = blockIdx.x * 4 + wave;
    if (tile >= ntiles) return;
    const int tbase = tile * 16;

    // ---- stage A = [h | agg] rows ----
    {
        const int node = tbase + m;
        const float4* hr = (const float4*)(h   + (size_t)node * HPREV);
        const float4* gr = (const float4*)(agg + (size_t)node * AGGD);
        float4* ar  = (float4*)&sA[wave][m * KP];
        float4* ar2 = (float4*)&sA[wave][m * KP + HPREV];
        if constexpr (HPREV == 64) {
            #pragma unroll
            for (int i = 0; i < 8; ++i) ar[kg * 8 + i] = hr[kg * 8 + i];
        } else {                          // HPREV == 8
            if (kg == 0) { ar[0] = hr[0]; ar[1] = hr[1]; }
        }
        if constexpr (AGGD == 64) {
            #pragma unroll
            for (int i = 0; i < 8; ++i) ar2[kg * 8 + i] = gr[kg * 8 + i];
        } else {                          // AGGD == 4
            if (kg == 0) ar2[0] = gr[0];
        }
    }
    asm volatile("s_wait_dscnt 0" ::: "memory");

    v8f acc[NC];
    const v8f zacc = {0.f, 0.f, 0.f, 0.f, 0.f, 0.f, 0.f, 0.f};
    #pragma unroll
    for (int nc = 0; nc < NC; ++nc) acc[nc] = zacc;

    #pragma unroll 4
    for (int kc = 0; kc < KC; ++kc) {
        const int kk = kc * 4 + kg * 2;
        v2f a = *(const v2f*)&sA[wave][m * KP + kk];
        #pragma unroll
        for (int nc = 0; nc < NC; ++nc) {
            v2f b = sW[(kk >> 1) * NP + nc * 16 + m];
            acc[nc] = __builtin_amdgcn_wmma_f32_16x16x4_f32(
                false, a, false, b, (short)0, acc[nc], false, false);
        }
    }

    #pragma unroll
    for (int nc = 0; nc < NC; ++nc) {
        const int n  = nc * 16 + m;
        const float bv = (n < NOUT) ? bias[n] : 0.f;
        #pragma unroll
        for (int v = 0; v < 8; ++v) {
            const int row  = v + kg * 8;
            const int node = tbase + row;
            float val = fmaxf(acc[nc][v] + bv, 0.f);
            if (n < NOUT) out[(size_t)node * NOUT + n] = val;
        }
    }
}

// ---------------------------------------------------------------------------
// Elementwise helpers
// ---------------------------------------------------------------------------
__global__ void zero_kernel(float* __restrict__ p, size_t n)
{
    size_t i = (size_t)blockIdx.x * blockDim.x + threadIdx.x;
    if (i < n) p[i] = 0.f;
}

__global__ void combine_kernel(float* __restrict__ out,
                               const float* b0, const float* b1, const float* b2,
                               const float* b3, const float* b4, const float* b5,
                               const float* b6,
                               const float* __restrict__ sw, int nbuf, size_t n)
{
    size_t i = (size_t)blockIdx.x * blockDim.x + threadIdx.x;
    if (i >= n) return;
    const float* bufs[7] = {b0, b1, b2, b3, b4, b5, b6};
    float acc = 0.f;
    for (int j = 0; j < nbuf; ++j) acc += sw[j] * bufs[j][i];
    out[i] = acc;
}

// ---------------------------------------------------------------------------
// Host schedule
// ---------------------------------------------------------------------------
extern "C" void kernel_launch(void* const* d_in, const int* in_sizes, int n_in,
                              void* d_out, int out_size, void* d_ws, size_t ws_size,
                              hipStream_t stream)
{
    const float*      x       = (const float*)d_in[0];
    const long long*  eidx    = (const long long*)d_in[1];
    const float*      ea      = (const float*)d_in[2];
    const float*      Wm1     = (const float*)d_in[3];
    const float*      bm1     = (const float*)d_in[4];
    const float*      Wu1     = (const float*)d_in[5];
    const float*      bu1     = (const float*)d_in[6];
    const float*      Wm_mid  = (const float*)d_in[7];   // [5][68][64]
    const float*      bm_mid  = (const float*)d_in[8];   // [5][64]
    const float*      Wu_mid  = (const float*)d_in[9];   // [5][128][64]
    const float*      bu_mid  = (const float*)d_in[10];  // [5][64]
    const float*      Wm_last = (const float*)d_in[11];  // [68][4]
    const float*      bm_last = (const float*)d_in[12];
    const float*      Wu_last = (const float*)d_in[13];  // [68][4]
    const float*      bu_last = (const float*)d_in[14];
    const float*      sw      = (const float*)d_in[15];  // [27]

    const int N = in_sizes[0] / 8;
    const int E = in_sizes[2] / 4;
    const long long* src = eidx;
    const long long* dst = eidx + E;

    const size_t NH = (size_t)N * 64;
    float* ws   = (float*)d_ws;
    float* x1   = ws + 0 * NH;
    float* x2w  = ws + 1 * NH;
    float* x3w  = ws + 2 * NH;
    float* x4w  = ws + 3 * NH;
    float* x5w  = ws + 4 * NH;
    float* x6w  = ws + 5 * NH;
    float* x7w  = ws + 6 * NH;
    float* t    = ws + 7 * NH;   // current unweighted layer output
    float* agg  = ws + 8 * NH;   // [N][64]
    float* agg4 = ws + 9 * NH;   // [N][4]

    const int etiles  = (E + 15) / 16;
    const int eblocks = (etiles + 3) / 4;
    const int ntiles  = (N + 15) / 16;
    const int nblocks = (ntiles + 3) / 4;

    auto zero = [&](float* p, size_t n) {
        zero_kernel<<<(unsigned)((n + 255) / 256), 256, 0, stream>>>(p, n);
    };

    // graph layer with HID-sized hidden state, weights Wm_mid[wi]/Wu_mid[wi]
    auto midLayer = [&](const float* hin, int wi, float* xout) {
        zero(agg, NH);
        msg_kernel<64, 64><<<eblocks, 128, 0, stream>>>(
            hin, ea, src, dst,
            Wm_mid + (size_t)wi * 68 * 64, bm_mid + (size_t)wi * 64, agg, etiles);
        upd_kernel<64, 64, 64><<<nblocks, 128, 0, stream>>>(
            hin, agg,
            Wu_mid + (size_t)wi * 128 * 64, bu_mid + (size_t)wi * 64, xout, ntiles);
    };

    const unsigned cblocks = (unsigned)((NH + 255) / 256);

    // ---- layer 1: x (N x 8) -> x1 ----
    zero(agg, NH);
    msg_kernel<8, 64><<<eblocks, 128, 0, stream>>>(x, ea, src, dst, Wm1, bm1, agg, etiles);
    upd_kernel<8, 64, 64><<<nblocks, 128, 0, stream>>>(x, agg, Wu1, bu1, x1, ntiles);

    // ---- layer 2 ----
    midLayer(x1, 0, t);
    combine_kernel<<<cblocks, 256, 0, stream>>>(x2w, x1, t, nullptr, nullptr, nullptr,
                                                nullptr, nullptr, sw + 0, 2, NH);
    // ---- layer 3 ----
    midLayer(x2w, 1, t);
    combine_kernel<<<cblocks, 256, 0, stream>>>(x3w, x1, x2w, t, nullptr, nullptr,
                                                nullptr, nullptr, sw + 2, 3, NH);
    // ---- layer 4 ----
    midLayer(x3w, 2, t);
    combine_kernel<<<cblocks, 256, 0, stream>>>(x4w, x1, x2w, x3w, t, nullptr,
                                                nullptr, nullptr, sw + 5, 4, NH);
    // ---- layer 5 (reference bug: reuses layer-4 weights) ----
    midLayer(x4w, 2, t);
    combine_kernel<<<cblocks, 256, 0, stream>>>(x5w, x1, x2w, x3w, x4w, t,
                                                nullptr, nullptr, sw + 9, 5, NH);
    // ---- layer 6 ----
    midLayer(x5w, 3, t);
    combine_kernel<<<cblocks, 256, 0, stream>>>(x6w, x1, x2w, x3w, x4w, x5w,
                                                t, nullptr, sw + 14, 6, NH);
    // ---- layer 7 ----
    midLayer(x6w, 4, t);
    combine_kernel<<<cblocks, 256, 0, stream>>>(x7w, x1, x2w, x3w, x4w, x5w,
                                                x6w, t, sw + 20, 7, NH);

    // ---- last layer: x7w -> out (N x 4) ----
    zero(agg4, (size_t)N * 4);
    msg_kernel<64, 4><<<eblocks, 128, 0, stream>>>(
        x7w, ea, src, dst, Wm_last, bm_last, agg4, etiles);
    upd_kernel<64, 4, 4><<<nblocks, 128, 0, stream>>>(
        x7w, agg4, Wu_last, bu_last, (float*)d_out, ntiles);
}